// SDPAttention_49941879717985
// MI455X (gfx1250) — compile-verified
//
#include <hip/hip_runtime.h>

#define KEY_DIM 128
#define SCORE_SCALE 0.08838834764831845f   // 1/sqrt(128)
#define ROW_STRIDE 132                     // 128 + 4 pad floats -> conflict-free LDS

typedef float v2f __attribute__((ext_vector_type(2)));
typedef float v4f __attribute__((ext_vector_type(4)));
typedef float v8f __attribute__((ext_vector_type(8)));

__device__ __forceinline__ int lower_bound_i32(const int* __restrict__ idx, int n, int key) {
    int lo = 0, hi = n;
    while (lo < hi) {
        int mid = (lo + hi) >> 1;
        if (idx[mid] < key) lo = mid + 1; else hi = mid;
    }
    return lo;
}

__device__ __forceinline__ float bcast_lane(float v, int srcLane) {
    return __uint_as_float(__builtin_amdgcn_readlane(__float_as_uint(v), srcLane));
}

// One 32-thread (single wave32) workgroup per graph. Fused single-pass
// online-softmax attention pooling over the graph's contiguous node range.
__global__ __launch_bounds__(32)
void seg_softmax_attn_kernel(const float* __restrict__ V,
                             const float* __restrict__ Q,
                             const int* __restrict__ bidx,
                             float* __restrict__ out,
                             int n_nodes) {
    __shared__ float qs[KEY_DIM];
    __shared__ float tile[16 * ROW_STRIDE];   // 16 x 128 V tile, padded rows
    __shared__ float sc[16];                  // per-tile scores (pre-scaled)

    const int g    = blockIdx.x;
    const int lane = threadIdx.x;

    // segment bounds via binary search on the sorted index array
    const int start = lower_bound_i32(bidx, n_nodes, g);
    const int end   = lower_bound_i32(bidx, n_nodes, g + 1);

    // stage Q[g] * (1/sqrt(d)) into LDS -> WMMA produces pre-scaled scores
    {
        const v4f* Q4 = (const v4f*)(Q + (size_t)g * KEY_DIM);
        *(v4f*)&qs[4 * lane] = Q4[lane] * SCORE_SCALE;
    }

    float m = -3.0e38f;   // running max (finite sentinel: exp underflows cleanly)
    float l = 0.0f;       // running denom
    v4f   acc = {};       // running weighted sum: dims [4*lane, 4*lane+4)

    // WMMA A-operand fetch pattern for V_WMMA_F32_16X16X4_F32:
    // lane (0..15) -> row = lane,    holds A[row, 4k+0], A[row, 4k+1]
    // lane (16..31)-> row = lane-16, holds A[row, 4k+2], A[row, 4k+3]
    const int row  = lane & 15;
    const int hi2  = (lane >> 4) << 1;          // 0 or 2
    const float* tline = &tile[row * ROW_STRIDE + hi2];
    const float* qline = &qs[hi2];

    for (int t = start; t < end; t += 16) {
        int cnt = end - t; if (cnt > 16) cnt = 16;

        __syncthreads();   // previous tile fully consumed before overwrite

        // stage 16x128 tile (zero-padded tail), coalesced 512B per row
        for (int r = 0; r < 16; ++r) {
            v4f val = {};
            if (r < cnt)
                val = *(const v4f*)(V + (size_t)(t + r) * KEY_DIM + 4 * lane);
            *(v4f*)&tile[r * ROW_STRIDE + 4 * lane] = val;
        }
        __syncthreads();

        // scores(16) = Vtile(16x128) . q_scaled(128); B = q broadcast across
        // all 16 columns, so every column of D equals the score vector.
        v8f c = {};
        #pragma unroll 8
        for (int k = 0; k < 32; ++k) {
            v2f a = *(const v2f*)(tline + 4 * k);
            v2f b = *(const v2f*)(qline + 4 * k);
            c = __builtin_amdgcn_wmma_f32_16x16x4_f32(
                    false, a, false, b, (short)0, c, false, false);
        }

        // D layout: vgpr r, lanes 0-15 -> M=r, lanes 16-31 -> M=r+8.
        // Lanes 0 and 16 publish rows 0-7 and 8-15.
        if ((lane & 15) == 0) {
            const int base = (lane >> 4) << 3;
            #pragma unroll
            for (int j = 0; j < 8; ++j) sc[base + j] = c[j];
        }
        __syncthreads();

        // ---- lane-parallel online softmax update ----
        // lane (i = lane&15) owns score i; lanes 16-31 hold duplicates.
        const float s = (row < cnt) ? sc[row] : -3.0e38f;

        // tile max via 4-step butterfly inside each 16-lane group
        float mt = s;
        #pragma unroll
        for (int d = 1; d < 16; d <<= 1)
            mt = fmaxf(mt, __shfl_xor(mt, d, 32));

        const float m_new = fmaxf(m, mt);
        float e = (row < cnt) ? __expf(s - m_new) : 0.0f;

        // tile sum of e via 4-step butterfly
        float es = e;
        #pragma unroll
        for (int d = 1; d < 16; d <<= 1)
            es += __shfl_xor(es, d, 32);

        const float rescale = __expf(m - m_new);
        m = m_new;
        l = l * rescale + es;
        acc *= rescale;

        // weighted accumulate: e_i broadcast via v_readlane (SGPR fma operand)
        for (int i = 0; i < cnt; ++i) {
            const float ei = bcast_lane(e, i);
            const v4f v = *(const v4f*)&tile[i * ROW_STRIDE + 4 * lane];
            acc += ei * v;
        }
    }

    const float invl = (l > 0.0f) ? (1.0f / l) : 0.0f;  // empty segment -> zeros
    v4f res = acc * invl;
    *(v4f*)(out + (size_t)g * KEY_DIM + 4 * lane) = res;
}

extern "C" void kernel_launch(void* const* d_in, const int* in_sizes, int n_in,
                              void* d_out, int out_size, void* d_ws, size_t ws_size,
                              hipStream_t stream) {
    const float* V    = (const float*)d_in[0];
    const float* Q    = (const float*)d_in[1];
    const int*   bidx = (const int*)d_in[2];
    float*       out  = (float*)d_out;

    const int n_nodes  = in_sizes[2];
    const int n_graphs = in_sizes[1] / KEY_DIM;

    seg_softmax_attn_kernel<<<n_graphs, 32, 0, stream>>>(V, Q, bidx, out, n_nodes);
}